// PointGNNConv_67611375173917
// MI455X (gfx1250) — compile-verified
//
#include <hip/hip_runtime.h>

// PointGNNConv optimized for MI455X (gfx1250, wave32, WMMA).
// e@Wf1 = rel@Wf1[:3,:] + x[src]@Wf1[3:,:] -> edge pass is a gather + rank-3
// update + atomic scatter (memory bound, ~655MB @ 23.3TB/s); all GEMMs are
// node-side N x 128 x 128 with v_wmma_f32_16x16x4_f32 at full fp32 precision.
// B-matrix is staged in LDS pair-swizzled (stride 288 dwords) so each WMMA B
// operand is one bank-conflict-free ds_load_b64 into an even-aligned pair.

#define NN    40000
#define DD    128
#define EE    640000
#define TILES 2500          // NN / 16, exact
#define SLOPE 0.01f
#define PSTR  288           // dwords per k-pair row in swizzled W (256 + 32 pad)
#define WSWZ  (64 * PSTR)   // floats per swizzled 128x128 weight matrix (18432)

typedef __attribute__((ext_vector_type(2))) float v2f;
typedef __attribute__((ext_vector_type(8))) float v8f;

__device__ __forceinline__ v8f wmma4(v2f a, v2f b, v8f c) {
  // (neg_a, A, neg_b, B, c_mod, C, reuse_a, reuse_b)
  return __builtin_amdgcn_wmma_f32_16x16x4_f32(false, a, false, b, (short)0, c,
                                               false, false);
}

__device__ __forceinline__ void wait_async0() {
#if __has_builtin(__builtin_amdgcn_s_wait_asynccnt)
  __builtin_amdgcn_s_wait_asynccnt(0);
#else
  asm volatile("s_wait_asynccnt 0x0" ::: "memory");
#endif
}

// Stage a row-major 128x128 weight matrix from global into pair-swizzled LDS:
// sW[(r/2)*PSTR + c*2 + (r&1)] = W[r][c].  256-thread cooperative copy.
__device__ __forceinline__ void stage_w_swizzled(const float* __restrict__ W,
                                                 float* __restrict__ sW) {
  const int tid = threadIdx.x;
  for (int i = tid * 4; i < DD * DD; i += 256 * 4) {
    const float4 w = *(const float4*)(W + i);
    const int r = i >> 7;       // row
    const int c = i & 127;      // col (multiple of 4)
    float* d = sW + (r >> 1) * PSTR + (c << 1) + (r & 1);
    d[0] = w.x; d[2] = w.y; d[4] = w.z; d[6] = w.w;
  }
}

// One wave computes a 16x128 tile: out = act(A[16x128] @ W + bias) (+resid)
// A: row-major stride 128 (global or LDS). sWsw: pair-swizzled W in LDS.
__device__ __forceinline__ void wave_gemm_16x128(const float* __restrict__ A,
                                                 const float* __restrict__ sWsw,
                                                 const float* __restrict__ bias,
                                                 const float* __restrict__ resid,
                                                 float* __restrict__ out,
                                                 int act) {
  const int lane = threadIdx.x & 31;
  const int half = lane >> 4;     // A layout: half 0 -> K,K+1 ; half 1 -> K+2,K+3
  const int l16  = lane & 15;

  v8f acc[8] = {};                              // 8 col tiles of 16 -> 128 cols
  const float* arow  = A + l16 * DD + 2 * half; // per-lane A row slice
  const float* bbase = sWsw + half * PSTR + l16 * 2;

  for (int k = 0; k < DD; k += 4) {
    const v2f a = *(const v2f*)(arow + k);      // global_load_b64
    const float* bk = bbase + k * (PSTR / 2);   // k/2 pair rows
#pragma unroll
    for (int ct = 0; ct < 8; ++ct) {
      const v2f b = *(const v2f*)(bk + ct * 32);  // ds_load_b64, conflict-free
      acc[ct] = wmma4(a, b, acc[ct]);
    }
  }

  // D layout: vgpr v: lanes 0-15 -> row v, lanes 16-31 -> row v+8; col = l16
#pragma unroll
  for (int ct = 0; ct < 8; ++ct) {
    const int col = ct * 16 + l16;
    const float bv = bias[col];
#pragma unroll
    for (int v = 0; v < 8; ++v) {
      const int row = v + 8 * half;
      float val = acc[ct][v] + bv;
      if (act) val = (val >= 0.f) ? val : SLOPE * val;
      if (resid) val += resid[row * DD + col];
      out[row * DD + col] = val;
    }
  }
}

// ---------------------------------------------------------------- kernel 1
// h1 = leaky(x@Wh1 + bh1), xs = x@Wf1[3:,:] + bf1   (both N x 128)
__global__ void node_pre(const float* __restrict__ x,
                         const float* __restrict__ Wh1,
                         const float* __restrict__ bh1,
                         const float* __restrict__ Wf1,
                         const float* __restrict__ bf1,
                         float* __restrict__ h1, float* __restrict__ xs) {
  extern __shared__ float smem[];
  float* sW0 = smem;            // swizzled Wh1        : WSWZ floats
  float* sW1 = smem + WSWZ;     // swizzled Wf1[3:,:]  : WSWZ floats

  stage_w_swizzled(Wh1, sW0);
  stage_w_swizzled(Wf1 + 3 * DD, sW1);   // skip the 3 rel rows
  __syncthreads();

  const int tile = blockIdx.x * 8 + (threadIdx.x >> 5);   // wave-uniform
  if (tile < TILES) {
    const float* A = x + (size_t)tile * 16 * DD;
    wave_gemm_16x128(A, sW0, bh1, nullptr, h1 + (size_t)tile * 16 * DD, 1);
    wave_gemm_16x128(A, sW1, bf1, nullptr, xs + (size_t)tile * 16 * DD, 0);
  }
}

// ---------------------------------------------------------------- kernel 2
// q[i] = tanh(h1[i] @ Wh2 + bh2) - pos[i]   (N x 3)
__global__ void delta_q(const float* __restrict__ h1,
                        const float* __restrict__ Wh2,
                        const float* __restrict__ bh2,
                        const float* __restrict__ pos,
                        float* __restrict__ q) {
  const int wave = threadIdx.x >> 5;
  const int lane = threadIdx.x & 31;
  const int i = blockIdx.x * 8 + wave;             // wave-uniform
  if (i >= NN) return;

  const float4 h = ((const float4*)(h1 + (size_t)i * DD))[lane];
  const int k0 = lane * 4;
  float s0 = 0.f, s1 = 0.f, s2 = 0.f;
  const float hv[4] = {h.x, h.y, h.z, h.w};
#pragma unroll
  for (int t = 0; t < 4; ++t) {
    const int k = k0 + t;
    s0 += hv[t] * Wh2[k * 3 + 0];
    s1 += hv[t] * Wh2[k * 3 + 1];
    s2 += hv[t] * Wh2[k * 3 + 2];
  }
#pragma unroll
  for (int off = 16; off; off >>= 1) {
    s0 += __shfl_xor(s0, off, 32);
    s1 += __shfl_xor(s1, off, 32);
    s2 += __shfl_xor(s2, off, 32);
  }
  if (lane < 3) {
    const float s = (lane == 0) ? s0 : (lane == 1) ? s1 : s2;
    q[i * 3 + lane] = tanhf(s + bh2[lane]) - pos[i * 3 + lane];
  }
}

// ---------------------------------------------------------------- kernel 3
// per edge: rel = pos[src] + q[dst]; m = leaky(xs[src] + rel @ Wf1[:3,:]);
// aggr[dst] += m   (global_atomic_add_f32, no return)
__global__ void edge_msg(const long long* __restrict__ ei,
                         const float* __restrict__ pos,
                         const float* __restrict__ q,
                         const float* __restrict__ xs,
                         const float* __restrict__ Wf1,
                         float* __restrict__ aggr) {
  __shared__ __align__(16) float sW[3 * DD];
  const int tid = threadIdx.x;

  // Stage Wf1 rows 0..2 (1.5KB) with CDNA5 async global->LDS DMA.
  // 96 threads = 3 full waves -> EXEC all-ones in every issuing wave.
  if (tid < 96) {
    const unsigned lds_off = (unsigned)(unsigned long long)(sW + tid * 4);
    const float* gsrc = Wf1 + tid * 4;
    asm volatile("global_load_async_to_lds_b128 %0, %1, off"
                 :: "v"(lds_off), "v"(gsrc) : "memory");
  }
  wait_async0();
  __syncthreads();

  const int wave = tid >> 5;
  const int lane = tid & 31;
  const long long e = (long long)blockIdx.x * 8 + wave;    // wave-uniform
  if (e >= EE) return;

  const int src = (int)ei[e];
  const int dst = (int)ei[EE + e];

  float r = 0.f;
  if (lane < 3) r = pos[src * 3 + lane] + q[dst * 3 + lane];
  const float r0 = __shfl(r, 0, 32);
  const float r1 = __shfl(r, 1, 32);
  const float r2 = __shfl(r, 2, 32);

  const int fb = lane * 4;
  const float4 xv = *(const float4*)(xs + (size_t)src * DD + fb);
  const float m[4] = {xv.x, xv.y, xv.z, xv.w};
  float* dsta = aggr + (size_t)dst * DD + fb;
#pragma unroll
  for (int j = 0; j < 4; ++j) {
    float v = m[j] + r0 * sW[fb + j] + r1 * sW[DD + fb + j] + r2 * sW[2 * DD + fb + j];
    v = (v >= 0.f) ? v : SLOPE * v;
    atomicAdd(dsta + j, v);
  }
}

// ---------------------------------------------------------------- kernel 4
// out = x + leaky(aggr@Wg1 + bg1) @ Wg2 + bg2
__global__ void node_out(const float* __restrict__ aggr,
                         const float* __restrict__ Wg1,
                         const float* __restrict__ bg1,
                         const float* __restrict__ Wg2,
                         const float* __restrict__ bg2,
                         const float* __restrict__ x,
                         float* __restrict__ out) {
  extern __shared__ float smem[];
  float* sW1 = smem;                  // swizzled Wg1 : WSWZ floats
  float* sW2 = smem + WSWZ;           // swizzled Wg2 : WSWZ floats
  float* sT  = smem + 2 * WSWZ;       // 8 waves * 2048 floats (t tiles)

  stage_w_swizzled(Wg1, sW1);
  stage_w_swizzled(Wg2, sW2);
  __syncthreads();

  const int wave = threadIdx.x >> 5;
  const int tile = blockIdx.x * 8 + wave;          // wave-uniform
  const bool valid = tile < TILES;
  float* tbuf = sT + wave * (16 * DD);

  if (valid)
    wave_gemm_16x128(aggr + (size_t)tile * 16 * DD, sW1, bg1, nullptr, tbuf, 1);
  __syncthreads();   // t tile visible in LDS before re-use as A operand
  if (valid)
    wave_gemm_16x128(tbuf, sW2, bg2, x + (size_t)tile * 16 * DD,
                     out + (size_t)tile * 16 * DD, 0);
}

// ---------------------------------------------------------------- launcher
extern "C" void kernel_launch(void* const* d_in, const int* in_sizes, int n_in,
                              void* d_out, int out_size, void* d_ws, size_t ws_size,
                              hipStream_t stream) {
  const float*     x   = (const float*)d_in[0];
  const float*     pos = (const float*)d_in[1];
  const long long* ei  = (const long long*)d_in[2];
  const float*     Wh1 = (const float*)d_in[3];
  const float*     bh1 = (const float*)d_in[4];
  const float*     Wh2 = (const float*)d_in[5];
  const float*     bh2 = (const float*)d_in[6];
  const float*     Wf1 = (const float*)d_in[7];
  const float*     bf1 = (const float*)d_in[8];
  const float*     Wg1 = (const float*)d_in[9];
  const float*     bg1 = (const float*)d_in[10];
  const float*     Wg2 = (const float*)d_in[11];
  const float*     bg2 = (const float*)d_in[12];
  float* out = (float*)d_out;

  float* h1   = (float*)d_ws;                 // N*128
  float* xs   = h1 + (size_t)NN * DD;         // N*128
  float* aggr = xs + (size_t)NN * DD;         // N*128
  float* q    = aggr + (size_t)NN * DD;       // N*3

  const int lds_pre = 2 * WSWZ * (int)sizeof(float);                 // 147456 B
  const int lds_out = (2 * WSWZ + 8 * 16 * DD) * (int)sizeof(float); // 212992 B
  hipFuncSetAttribute((const void*)node_pre,
                      hipFuncAttributeMaxDynamicSharedMemorySize, lds_pre);
  hipFuncSetAttribute((const void*)node_out,
                      hipFuncAttributeMaxDynamicSharedMemorySize, lds_out);

  hipMemsetAsync(aggr, 0, (size_t)NN * DD * sizeof(float), stream);

  node_pre<<<(TILES + 7) / 8, 256, lds_pre, stream>>>(x, Wh1, bh1, Wf1, bf1, h1, xs);
  delta_q<<<NN / 8, 256, 0, stream>>>(h1, Wh2, bh2, pos, q);
  edge_msg<<<EE / 8, 256, 0, stream>>>(ei, pos, q, xs, Wf1, aggr);
  node_out<<<(TILES + 7) / 8, 256, lds_out, stream>>>(aggr, Wg1, bg1, Wg2, bg2, x, out);
}